// RayNetwork_82901458747595
// MI455X (gfx1250) — compile-verified
//
#include <hip/hip_runtime.h>

#define BB   2
#define NN   8192
#define MM   8192
#define FF   128
#define OUTD 512
#define KNN  10

typedef float v2f_t __attribute__((ext_vector_type(2)));
typedef float v8f_t __attribute__((ext_vector_type(8)));

// ---------------------------------------------------------------------------
// Kernel 1: brute-force KNN (top-10 smallest distances) per query point.
// One thread per query; candidate point-cloud tiles staged in LDS (all lanes
// read the same candidate per step -> LDS broadcast, conflict-free).
// Top-10 kept sorted in registers: replace max, one unrolled bubble pass.
// ---------------------------------------------------------------------------
__global__ __launch_bounds__(256)
void knn_kernel(const float* __restrict__ geometry,
                const float* __restrict__ pcd,
                int*   __restrict__ knn_idx,
                float* __restrict__ knn_w) {
  const int TM = 2048;                 // 2048 * 16B = 32 KB LDS
  __shared__ float4 sp[TM];

  const int tid = threadIdx.x;
  const int row = blockIdx.x * 256 + tid;      // global query id in [0, B*N)
  const int b   = row / NN;                    // uniform across the block

  const float* g = geometry + (size_t)row * 3;
  const float qx = g[0], qy = g[1], qz = g[2];

  float bd[KNN];
  int   bi[KNN];
#pragma unroll
  for (int i = 0; i < KNN; ++i) { bd[i] = 3.4e38f; bi[i] = 0; }

  const float* pb = pcd + (size_t)b * MM * 3;
  for (int t = 0; t < MM; t += TM) {
    // cooperative tile load into LDS as padded float4
    for (int i = tid; i < TM; i += 256) {
      const float* p = pb + (size_t)(t + i) * 3;
      sp[i] = make_float4(p[0], p[1], p[2], 0.0f);
    }
    __syncthreads();

    for (int j = 0; j < TM; ++j) {
      const float4 p = sp[j];
      const float dx = qx - p.x;
      const float dy = qy - p.y;
      const float dz = qz - p.z;
      const float d2 = dx * dx + dy * dy + dz * dz;
      if (d2 < bd[KNN - 1]) {
        bd[KNN - 1] = d2;
        bi[KNN - 1] = t + j;
        // one bubble pass restores sorted order (only last elem out of place)
#pragma unroll
        for (int q = KNN - 1; q > 0; --q) {
          if (bd[q] < bd[q - 1]) {
            float td = bd[q]; bd[q] = bd[q - 1]; bd[q - 1] = td;
            int   ti = bi[q]; bi[q] = bi[q - 1]; bi[q - 1] = ti;
          }
        }
      }
    }
    __syncthreads();
  }

  // inverse-distance weights, normalized (matches reference: sqrt + 1e-8)
  float w[KNN];
  float s = 0.0f;
#pragma unroll
  for (int i = 0; i < KNN; ++i) {
    const float d = sqrtf(fmaxf(bd[i], 0.0f));
    w[i] = 1.0f / (d + 1e-8f);
    s += w[i];
  }
  const float inv = 1.0f / s;
#pragma unroll
  for (int i = 0; i < KNN; ++i) {
    knn_idx[(size_t)row * KNN + i] = bi[i];
    knn_w  [(size_t)row * KNN + i] = w[i] * inv;
  }
}

// ---------------------------------------------------------------------------
// Kernel 2: weighted feature gather  interp[row][0:128] = sum_k w*feat[idx]
// One wave32 per row; lane owns 4 consecutive floats (128 = 32*4).
// ---------------------------------------------------------------------------
__global__ __launch_bounds__(256)
void interp_kernel(const float* __restrict__ features,
                   const int*   __restrict__ knn_idx,
                   const float* __restrict__ knn_w,
                   float*       __restrict__ interp) {
  const int lane = threadIdx.x & 31;
  const int wid  = threadIdx.x >> 5;
  const int row  = blockIdx.x * 8 + wid;
  const int b    = row / NN;

  const float4* fbase = (const float4*)(features + (size_t)b * MM * FF);
  float4 acc = make_float4(0.f, 0.f, 0.f, 0.f);
#pragma unroll
  for (int k = 0; k < KNN; ++k) {
    const int   id = knn_idx[(size_t)row * KNN + k];
    const float w  = knn_w [(size_t)row * KNN + k];
    const float4 f = fbase[(size_t)id * (FF / 4) + lane];
    acc.x += w * f.x; acc.y += w * f.y; acc.z += w * f.z; acc.w += w * f.w;
  }
  ((float4*)(interp + (size_t)row * FF))[lane] = acc;
}

// ---------------------------------------------------------------------------
// Kernel 3: projection GEMM  out[16384x512] = interp[16384x128] * W^T + bias
// Full-f32 WMMA: V_WMMA_F32_16X16X4_F32, one 16x16 tile per wave, K-loop x32.
// A layout (16x4 f32): lane L, vgpr v -> (M = L%16, K = v + 2*(L/16))
// B layout (4x16 f32): lane L, vgpr v -> (K = v + 2*(L/16), N = L%16)
// C/D layout:          lane L, vgpr v -> (M = v + 8*(L/16), N = L%16)
// ---------------------------------------------------------------------------
__global__ __launch_bounds__(256)
void proj_wmma_kernel(const float* __restrict__ interp,
                      const float* __restrict__ W,     // [OUTD][FF] row-major
                      const float* __restrict__ bias,  // [OUTD]
                      float*       __restrict__ out) { // [B*N][OUTD]
  const int lane = threadIdx.x & 31;
  const int wid  = threadIdx.x >> 5;
  const int tile = blockIdx.x * 8 + wid;

  const int col_tiles = OUTD / 16;         // 32
  const int rt = tile / col_tiles;
  const int ct = tile % col_tiles;
  const int row0 = rt * 16;
  const int n0   = ct * 16;

  const int m  = lane & 15;                // M for A, N for B
  const int kh = lane >> 4;                // K-half select

  const float* arow = interp + (size_t)(row0 + m) * FF + 2 * kh;
  const float* brow = W      + (size_t)(n0   + m) * FF + 2 * kh;

  v8f_t c = {};
#pragma unroll
  for (int kk = 0; kk < FF; kk += 4) {
    const v2f_t a  = *(const v2f_t*)(arow + kk);
    const v2f_t bv = *(const v2f_t*)(brow + kk);
    c = __builtin_amdgcn_wmma_f32_16x16x4_f32(
        /*neg_a=*/false, a, /*neg_b=*/false, bv,
        /*c_mod=*/(short)0, c, /*reuse_a=*/false, /*reuse_b=*/false);
  }

  const float bsc = bias[n0 + m];          // column = lane%16
#pragma unroll
  for (int v = 0; v < 8; ++v) {
    const int mr = v + 8 * kh;
    out[(size_t)(row0 + mr) * OUTD + n0 + m] = c[v] + bsc;
  }
}

// ---------------------------------------------------------------------------
extern "C" void kernel_launch(void* const* d_in, const int* in_sizes, int n_in,
                              void* d_out, int out_size, void* d_ws, size_t ws_size,
                              hipStream_t stream) {
  (void)in_sizes; (void)n_in; (void)out_size; (void)ws_size;

  const float* geometry = (const float*)d_in[0];   // [B,N,3]
  const float* pcd      = (const float*)d_in[1];   // [B,M,3]
  const float* features = (const float*)d_in[2];   // [B,M,F]
  const float* W        = (const float*)d_in[3];   // [OUT,F]
  const float* bias     = (const float*)d_in[4];   // [OUT]
  // d_in[5] = k (==10), baked in as KNN

  char* ws = (char*)d_ws;
  int*   knn_idx = (int*)ws;
  float* knn_w   = (float*)(ws + (size_t)BB * NN * KNN * sizeof(int));
  float* interp  = (float*)(ws + 2ull * BB * NN * KNN * sizeof(int));
  float* out     = (float*)d_out;

  knn_kernel<<<(BB * NN) / 256, 256, 0, stream>>>(geometry, pcd, knn_idx, knn_w);
  interp_kernel<<<(BB * NN) / 8, 256, 0, stream>>>(features, knn_idx, knn_w, interp);

  const int tiles = ((BB * NN) / 16) * (OUTD / 16);  // 32768 wave-tiles
  proj_wmma_kernel<<<tiles / 8, 256, 0, stream>>>(interp, W, bias, out);
}